// DecoderMultiHeadAttention_2671469658566
// MI455X (gfx1250) — compile-verified
//
#include <hip/hip_runtime.h>

// ---------------------------------------------------------------------------
// Decoder MHA on gfx1250 (MI455X), all-f32 path via V_WMMA_F32_16X16X4_F32.
// ---------------------------------------------------------------------------

typedef __attribute__((ext_vector_type(2))) float v2f;
typedef __attribute__((ext_vector_type(8))) float v8f;

namespace cfg {
constexpr int D_MODEL = 1024;
constexpr int N_HEAD  = 16;
constexpr int D_K     = 64;
constexpr int NB      = 2;      // batch
constexpr int SS      = 2048;   // sequence
constexpr int BM = 64, BN = 128, BK = 32;   // GEMM workgroup tile
}

// ===========================================================================
// GEMM: out = A[M,K] @ W[K,N] (+ bias). HEAD_SPLIT scatters the result into
// [B, H, S, D_K] layout for the attention stage.
// 8 waves arranged 2x4; each wave owns a 32x32 register tile (4 v8f accs),
// so each K-step is 2 A-frag + 2 B-frag ds_load_b64 feeding 4 WMMAs (1:1).
// WMMA f32 16x16x4 fragments:
//   A frag (16x4): lanes 0-15 hold (K=k, K=k+1) for row M=lane; lanes 16-31
//                  hold (K=k+2, K=k+3) for row M=lane-16.
//   B frag (4x16): v0 = B[k][n] (lanes 0-15) / B[k+2][n] (lanes 16-31),
//                  v1 = B[k+1][n] / B[k+3][n].
//   C/D (16x16):   VGPR r, lane L -> row r + 8*(L>=16), col L&15.
// ===========================================================================
template <bool HAS_BIAS, bool HEAD_SPLIT>
__global__ __launch_bounds__(256) void gemm_wmma_f32(
    const float* __restrict__ A, const float* __restrict__ W,
    const float* __restrict__ bias, float* __restrict__ out,
    int M, int K, int N) {
  using namespace cfg;
  __shared__ float As[BM][BK + 4];   // stride 36: 16B-aligned rows, conflict-free frag reads
  __shared__ float Bs[BK][BN + 8];   // stride 136: halves (rows k, k+2) hit disjoint banks

  const int tid  = threadIdx.x;
  const int wave = tid >> 5;
  const int lane = tid & 31;
  const int half = lane >> 4;
  const int lm   = lane & 15;

  const int m0 = blockIdx.y * BM;
  const int n0 = blockIdx.x * BN;
  const int mOff = (wave >> 2) * 32;  // 2 row-groups of 32
  const int nOff = (wave & 3) * 32;   // 4 col-groups of 32

  v8f acc00 = {}, acc01 = {}, acc10 = {}, acc11 = {};

  for (int k0 = 0; k0 < K; k0 += BK) {
    __syncthreads();
    // Cooperative tile loads: A 64x32 (512 float4), B 32x128 (1024 float4).
#pragma unroll
    for (int i = 0; i < 2; ++i) {
      int idx = tid + i * 256;
      int ar = idx >> 3, ac = (idx & 7) * 4;
      *(float4*)&As[ar][ac] = *(const float4*)&A[(size_t)(m0 + ar) * K + k0 + ac];
    }
#pragma unroll
    for (int i = 0; i < 4; ++i) {
      int idx = tid + i * 256;
      int br = idx >> 5, bc = (idx & 31) * 4;
      *(float4*)&Bs[br][bc] = *(const float4*)&W[(size_t)(k0 + br) * N + n0 + bc];
    }
    __syncthreads();

#pragma unroll
    for (int kk = 0; kk < BK; kk += 4) {
      const int ks = kk + 2 * half;
      v2f a0, a1, b0, b1;
      a0.x = As[mOff + lm][ks];
      a0.y = As[mOff + lm][ks + 1];
      a1.x = As[mOff + 16 + lm][ks];
      a1.y = As[mOff + 16 + lm][ks + 1];
      b0.x = Bs[ks][nOff + lm];
      b0.y = Bs[ks + 1][nOff + lm];
      b1.x = Bs[ks][nOff + 16 + lm];
      b1.y = Bs[ks + 1][nOff + 16 + lm];
      acc00 = __builtin_amdgcn_wmma_f32_16x16x4_f32(false, a0, false, b0,
                                                    (short)0, acc00, false, false);
      acc01 = __builtin_amdgcn_wmma_f32_16x16x4_f32(false, a0, false, b1,
                                                    (short)0, acc01, false, false);
      acc10 = __builtin_amdgcn_wmma_f32_16x16x4_f32(false, a1, false, b0,
                                                    (short)0, acc10, false, false);
      acc11 = __builtin_amdgcn_wmma_f32_16x16x4_f32(false, a1, false, b1,
                                                    (short)0, acc11, false, false);
    }
  }

  // Epilogue: C layout row = r + 8*half, col = lm (per 16x16 tile).
#pragma unroll
  for (int tr = 0; tr < 2; ++tr) {
    const int rowBase = m0 + mOff + tr * 16 + 8 * half;
#pragma unroll
    for (int r = 0; r < 8; ++r) {
      const int row = rowBase + r;
#pragma unroll
      for (int tc = 0; tc < 2; ++tc) {
        const int col = n0 + nOff + tc * 16 + lm;
        float val;
        if (tr == 0) val = (tc == 0) ? acc00[r] : acc01[r];
        else         val = (tc == 0) ? acc10[r] : acc11[r];
        if (HAS_BIAS) val += bias[col];
        if (HEAD_SPLIT) {
          const int bb = row >> 11;        // / SS
          const int ss = row & (SS - 1);
          const int h  = col >> 6;         // / D_K
          const int d  = col & (D_K - 1);
          out[(((size_t)bb * N_HEAD + h) * SS + ss) * D_K + d] = val;
        } else {
          out[(size_t)row * N + col] = val;
        }
      }
    }
  }
}

// ===========================================================================
// Flash-style attention. Block = 256 threads = 8 waves; each wave owns a
// 16-query block of one (b,h); the block streams 16-key K/V tiles via LDS.
// ===========================================================================
__global__ __launch_bounds__(256) void attn_wmma_f32(
    const float* __restrict__ Qh, const float* __restrict__ Kh,
    const float* __restrict__ Vh, const int* __restrict__ mask,
    float* __restrict__ O) {
  using namespace cfg;
  __shared__ float Ks[16][68];        // K tile, row = key, col = d
  __shared__ float Vs[16][72];        // V tile, row = key, col = d
  __shared__ float Ps[8][16][20];     // per-wave P (16 queries x 16 keys)

  const int tid  = threadIdx.x;
  const int wave = tid >> 5;
  const int lane = tid & 31;
  const int half = lane >> 4;
  const int lm   = lane & 15;

  constexpr int nQB = SS / 128;       // query blocks per (b,h)
  const int blk = blockIdx.x;
  const int bh  = blk / nQB;          // b*N_HEAD + h
  const int q0  = (blk % nQB) * 128;
  const int b   = bh >> 4;
  const int h   = bh & (N_HEAD - 1);
  const int qBase = q0 + wave * 16;

  const float* Qp = Qh + ((size_t)bh * SS + qBase) * D_K;
  const float* Kp = Kh + (size_t)bh * SS * D_K;
  const float* Vp = Vh + (size_t)bh * SS * D_K;
  const int*   Mp = mask + ((size_t)b * SS + qBase) * SS;

  // Q fragments (16x64 per wave): frag i covers K-dim [4i,4i+4).
  v2f qf[16];
  {
    const float* qrow = Qp + (size_t)lm * D_K + 2 * half;
#pragma unroll
    for (int i = 0; i < 16; ++i) {
      qf[i].x = qrow[4 * i];
      qf[i].y = qrow[4 * i + 1];
    }
  }

  v8f acc[4] = {{}, {}, {}, {}};
  float mi[8], li[8];
#pragma unroll
  for (int r = 0; r < 8; ++r) { mi[r] = -1e30f; li[r] = 0.0f; }

  const float inv_t = 0.125f;         // 1/sqrt(64)

  for (int j0 = 0; j0 < SS; j0 += 16) {
    __syncthreads();                  // previous tile fully consumed
    {                                 // cooperative K/V tile load (1 float4 each)
      const int row = tid >> 4, c4 = (tid & 15) * 4;
      *(float4*)&Ks[row][c4] = *(const float4*)&Kp[(size_t)(j0 + row) * D_K + c4];
      *(float4*)&Vs[row][c4] = *(const float4*)&Vp[(size_t)(j0 + row) * D_K + c4];
    }
    __syncthreads();

    // S = Q @ K^T  (16x16), K^T fragment: B[k][n] = Ks[n][k].
    v8f s = {};
#pragma unroll
    for (int i = 0; i < 16; ++i) {
      const int ks = 4 * i + 2 * half;
      v2f bf;
      bf.x = Ks[lm][ks];
      bf.y = Ks[lm][ks + 1];
      s = __builtin_amdgcn_wmma_f32_16x16x4_f32(false, qf[i], false, bf,
                                                (short)0, s, false, false);
    }

    // Mask + online softmax. Row r+8*half lives across one 16-lane half.
#pragma unroll
    for (int r = 0; r < 8; ++r) {
      const int qrow = r + 8 * half;
      const int msk  = Mp[(size_t)qrow * SS + j0 + lm];
      const float sc = msk ? s[r] * inv_t : -1e30f;
      float mx = sc;
      mx = fmaxf(mx, __shfl_xor(mx, 1, 32));
      mx = fmaxf(mx, __shfl_xor(mx, 2, 32));
      mx = fmaxf(mx, __shfl_xor(mx, 4, 32));
      mx = fmaxf(mx, __shfl_xor(mx, 8, 32));
      const float mnew = fmaxf(mi[r], mx);
      const float corr = __expf(mi[r] - mnew);
      const float p    = msk ? __expf(sc - mnew) : 0.0f;  // exact 0 like reference
      float rs = p;
      rs += __shfl_xor(rs, 1, 32);
      rs += __shfl_xor(rs, 2, 32);
      rs += __shfl_xor(rs, 4, 32);
      rs += __shfl_xor(rs, 8, 32);
      li[r] = li[r] * corr + rs;
      mi[r] = mnew;
#pragma unroll
      for (int t = 0; t < 4; ++t) acc[t][r] *= corr;
      Ps[wave][qrow][lm] = p;         // C-layout -> row-major staging
    }
    __syncthreads();                  // P visible cross-lane (and K reads done)

    // acc += P(16x16) @ V(16x64)
#pragma unroll
    for (int i = 0; i < 4; ++i) {
      const int ks = 4 * i + 2 * half;
      v2f a;
      a.x = Ps[wave][lm][ks];
      a.y = Ps[wave][lm][ks + 1];
#pragma unroll
      for (int t = 0; t < 4; ++t) {
        v2f bf;
        bf.x = Vs[ks][t * 16 + lm];
        bf.y = Vs[ks + 1][t * 16 + lm];
        acc[t] = __builtin_amdgcn_wmma_f32_16x16x4_f32(false, a, false, bf,
                                                       (short)0, acc[t], false, false);
      }
    }
  }

  // Normalize and write O in [B, S, D_MODEL] layout.
  float linv[8];
#pragma unroll
  for (int r = 0; r < 8; ++r) linv[r] = (li[r] > 0.0f) ? (1.0f / li[r]) : 0.0f;
  float* Op = O + ((size_t)b * SS + qBase) * D_MODEL + h * D_K;
#pragma unroll
  for (int r = 0; r < 8; ++r) {
    const int qrow = r + 8 * half;
#pragma unroll
    for (int t = 0; t < 4; ++t) {
      Op[(size_t)qrow * D_MODEL + t * 16 + lm] = acc[t][r] * linv[r];
    }
  }
}

// ===========================================================================
extern "C" void kernel_launch(void* const* d_in, const int* in_sizes, int n_in,
                              void* d_out, int out_size, void* d_ws, size_t ws_size,
                              hipStream_t stream) {
  using namespace cfg;
  (void)in_sizes; (void)n_in; (void)out_size; (void)ws_size;

  const float* q    = (const float*)d_in[0];
  const float* k    = (const float*)d_in[1];
  const float* v    = (const float*)d_in[2];
  const int*   mask = (const int*)d_in[3];
  const float* wq   = (const float*)d_in[4];
  const float* bq   = (const float*)d_in[5];
  const float* wk   = (const float*)d_in[6];
  const float* wv   = (const float*)d_in[7];
  const float* bv   = (const float*)d_in[8];
  const float* wf   = (const float*)d_in[9];
  const float* bf   = (const float*)d_in[10];
  float* out = (float*)d_out;

  const size_t tsz = (size_t)NB * SS * D_MODEL;   // elements per tensor
  float* Qh = (float*)d_ws;
  float* Kh = Qh + tsz;
  float* Vh = Kh + tsz;
  float* O  = Vh + tsz;

  const int M = NB * SS;                          // 4096
  dim3 gGrid(D_MODEL / BN, M / BM);               // (8, 64)
  dim3 blk(256);

  gemm_wmma_f32<true,  true ><<<gGrid, blk, 0, stream>>>(q, wq, bq,      Qh, M, D_MODEL, D_MODEL);
  gemm_wmma_f32<false, true ><<<gGrid, blk, 0, stream>>>(k, wk, nullptr, Kh, M, D_MODEL, D_MODEL);
  gemm_wmma_f32<true,  true ><<<gGrid, blk, 0, stream>>>(v, wv, bv,      Vh, M, D_MODEL, D_MODEL);

  attn_wmma_f32<<<dim3(NB * N_HEAD * (SS / 128)), blk, 0, stream>>>(Qh, Kh, Vh, mask, O);

  gemm_wmma_f32<true,  false><<<gGrid, blk, 0, stream>>>(O, wf, bf, out, M, D_MODEL, D_MODEL);
}